// DeformableAttentionModule_44504451121271
// MI455X (gfx1250) — compile-verified
//
#include <hip/hip_runtime.h>

typedef float v2f __attribute__((ext_vector_type(2)));
typedef float v8f __attribute__((ext_vector_type(8)));

#define B_     4
#define C_     256
#define H_     128
#define W_     128
#define NQ     16384   // H_*W_
#define HEADS_ 8
#define DHEAD  32
#define PTS    4

// fp32 WMMA GEMM, Out = A * Wt + bias (+resid), optional transposed store.
//   LDA, N compile-time so k-step addressing folds into load immediate offsets.
//   A_COL:  A[k*LDA + m]  (x feature-map layout, LDA = NQ)
//   !A_COL: A[m*LDA + k]  (row-major activations, LDA = K)
// Block: 256 threads = 8 waves; wave computes a 16(M) x 32(N) tile (2 WMMA acc),
// block covers 128 M x 32 N. K fixed at 256.
template<int LDA, int N, bool A_COL, bool TRANS_OUT, bool RESID>
__global__ __launch_bounds__(256) void gemm_wmma(
    const float* __restrict__ A, long aBatch,
    const float* __restrict__ Wt,
    const float* __restrict__ bias,
    const float* __restrict__ resid, long rBatch,
    float* __restrict__ Out, long oBatch, int M)
{
    constexpr int K = 256;
    __shared__ float tile[8][16][17];   // per-wave staging (TRANS_OUT only)

    const int lane = threadIdx.x & 31;
    const int wave = threadIdx.x >> 5;
    const int half = lane >> 4;         // 0: lanes 0-15, 1: lanes 16-31
    const int lm   = lane & 15;
    const int b    = blockIdx.z;
    const int n0   = blockIdx.x * 32;
    const int m0   = (blockIdx.y * 8 + wave) * 16;

    // Per-lane base pointers (computed once; loop uses immediate offsets).
    // A fragment (16x4): lane holds row m = lm, k = k0 + 2*half + {0,1}.
    const float* pA;
    if (A_COL) pA = A + (long)b * aBatch + (long)(2 * half) * LDA + (m0 + lm);
    else       pA = A + (long)b * aBatch + (long)(m0 + lm) * LDA + 2 * half;
    // B fragment (4x16): lane holds col n = lm, k = k0 + 2*half + {0,1}.
    const float* pB = Wt + (long)(2 * half) * N + (n0 + lm);

    const float bv0 = bias[n0 + lm];
    const float bv1 = bias[n0 + 16 + lm];
    v8f acc0, acc1;
    #pragma unroll
    for (int r = 0; r < 8; ++r) { acc0[r] = bv0; acc1[r] = bv1; }

    for (int g = 0; g < K / 32; ++g) {
        #pragma unroll
        for (int j = 0; j < 8; ++j) {          // k-step of 4 each
            v2f a, b0, b1;
            if (A_COL) {
                a.x = pA[(j * 4    ) * LDA];   // imm offsets, 64B-contig per half-wave
                a.y = pA[(j * 4 + 1) * LDA];
            } else {
                a = *(const v2f*)(pA + j * 4); // contiguous k-pair: one b64 load
            }
            b0.x = pB[(j * 4    ) * N];
            b0.y = pB[(j * 4 + 1) * N];
            b1.x = pB[(j * 4    ) * N + 16];
            b1.y = pB[(j * 4 + 1) * N + 16];

            acc0 = __builtin_amdgcn_wmma_f32_16x16x4_f32(
                false, a, false, b0, (short)0, acc0, false, false);
            acc1 = __builtin_amdgcn_wmma_f32_16x16x4_f32(
                false, a, false, b1, (short)0, acc1, false, false);
        }
        pA += A_COL ? (long)32 * LDA : 32;     // one bump per group of 8 k-steps
        pB += 32 * N;
    }

    if (!TRANS_OUT) {
        float* op = Out + (long)b * oBatch;
        #pragma unroll
        for (int r = 0; r < 8; ++r) {
            const int mm = m0 + r + 8 * half;  // C/D layout: M = r + 8*(lane/16)
            op[(long)mm * N + n0 + lm]      = acc0[r];
            op[(long)mm * N + n0 + 16 + lm] = acc1[r];
        }
    } else {
        float* op = Out + (long)b * oBatch;
        const float* rp = resid + (long)b * rBatch;
        #pragma unroll
        for (int nt = 0; nt < 2; ++nt) {
            const v8f acc = nt ? acc1 : acc0;
            #pragma unroll
            for (int r = 0; r < 8; ++r)
                tile[wave][r + 8 * half][lm] = acc[r];
            // per-wave tile; LDS ops from one wave stay in order
            #pragma unroll
            for (int j = 0; j < 8; ++j) {
                const int e  = j * 32 + lane;
                const int mm = e & 15;         // lanes 0-15 contiguous in m
                const int cc = e >> 4;
                float vv = tile[wave][mm][cc];
                const long idx = (long)(n0 + nt * 16 + cc) * M + (m0 + mm);
                if (RESID) vv += rp[idx];      // residual q == x at identical layout
                op[idx] = vv;
            }
        }
    }
}

// Deformable bilinear sampling + softmax attention combine.
// One thread per (b, n, head, d); 32 lanes of a wave span d=32 -> coalesced gathers.
__global__ __launch_bounds__(256) void deform_sample(
    const float* __restrict__ v,      // (B, Nq, 256)
    const float* __restrict__ off,    // (B, Nq, 64) = (heads, points, 2)
    const float* __restrict__ attnl,  // (B, Nq, 32) = (heads, points) logits
    float* __restrict__ outp)         // (B, Nq, 256)
{
    const long idx = (long)blockIdx.x * blockDim.x + threadIdx.x;
    const int  dd  = (int)(idx & (DHEAD - 1));
    const int  h   = (int)((idx >> 5) & (HEADS_ - 1));
    const long bn  = idx >> 8;                 // b*Nq + n
    const int  n   = (int)(bn & (NQ - 1));

    // softmax over the 4 points of this head
    const float* lg = attnl + bn * 32 + h * 4;
    float l0 = lg[0], l1 = lg[1], l2 = lg[2], l3 = lg[3];
    float mx = fmaxf(fmaxf(l0, l1), fmaxf(l2, l3));
    float e0 = __expf(l0 - mx), e1 = __expf(l1 - mx);
    float e2 = __expf(l2 - mx), e3 = __expf(l3 - mx);
    float inv = 1.0f / (e0 + e1 + e2 + e3);
    float aw[4] = { e0 * inv, e1 * inv, e2 * inv, e3 * inv };

    const int   iyr = n >> 7;
    const int   ixr = n & (W_ - 1);
    const float gx = (float)ixr * (1.0f / (float)(W_ - 1));
    const float gy = (float)iyr * (1.0f / (float)(H_ - 1));

    const float* vb  = v + (bn >> 14) * (long)NQ * C_;
    const float* ofp = off + bn * 64 + h * 8;

    float sum = 0.0f;
    #pragma unroll
    for (int p = 0; p < PTS; ++p) {
        const float lx = gx + ofp[2 * p + 0] * (1.0f / (float)W_);
        const float ly = gy + ofp[2 * p + 1] * (1.0f / (float)H_);
        const float fx = lx * (float)W_ - 0.5f;
        const float fy = ly * (float)H_ - 0.5f;
        const float x0f = floorf(fx), y0f = floorf(fy);
        const float wx = fx - x0f, wy = fy - y0f;
        const int x0 = (int)x0f, y0 = (int)y0f;

        float s = 0.0f;
        #pragma unroll
        for (int cy = 0; cy < 2; ++cy) {
            #pragma unroll
            for (int cx = 0; cx < 2; ++cx) {
                const int xi = x0 + cx, yi = y0 + cy;
                const bool ok = (xi >= 0) & (xi < W_) & (yi >= 0) & (yi < H_);
                const int xc = min(max(xi, 0), W_ - 1);
                const int yc = min(max(yi, 0), H_ - 1);
                const float g = vb[(long)(yc * W_ + xc) * C_ + h * DHEAD + dd];
                const float wgt = (cx ? wx : 1.0f - wx) * (cy ? wy : 1.0f - wy);
                s += ok ? g * wgt : 0.0f;
            }
        }
        sum = fmaf(aw[p], s, sum);
    }
    outp[bn * C_ + h * DHEAD + dd] = sum;
}

extern "C" void kernel_launch(void* const* d_in, const int* in_sizes, int n_in,
                              void* d_out, int out_size, void* d_ws, size_t ws_size,
                              hipStream_t stream)
{
    const float* x    = (const float*)d_in[0];
    const float* Wv   = (const float*)d_in[1];
    const float* bv   = (const float*)d_in[2];
    const float* Wo   = (const float*)d_in[3];
    const float* bo   = (const float*)d_in[4];
    const float* Wa   = (const float*)d_in[5];
    const float* ba   = (const float*)d_in[6];
    const float* Wout = (const float*)d_in[7];
    const float* bout = (const float*)d_in[8];
    float* out = (float*)d_out;

    char* ws = (char*)d_ws;
    float* v    = (float*)ws;  ws += (size_t)B_ * NQ * C_ * sizeof(float);   // 64 MB
    float* offb = (float*)ws;  ws += (size_t)B_ * NQ * 64 * sizeof(float);   // 16 MB
    float* attn = (float*)ws;  ws += (size_t)B_ * NQ * 32 * sizeof(float);   //  8 MB
    float* outp = (float*)ws;                                                // 64 MB

    dim3 blk(256);
    // 1) value projection: v = q @ W_value + b_value    (A = x, column-major in m)
    gemm_wmma<NQ, C_, true, false, false><<<dim3(C_ / 32, NQ / 128, B_), blk, 0, stream>>>(
        x, (long)C_ * NQ, Wv, bv, nullptr, 0, v, (long)NQ * C_, NQ);
    // 2) sampling offsets (N = 64)
    gemm_wmma<NQ, 64, true, false, false><<<dim3(64 / 32, NQ / 128, B_), blk, 0, stream>>>(
        x, (long)C_ * NQ, Wo, bo, nullptr, 0, offb, (long)NQ * 64, NQ);
    // 3) attention logits (N = 32)
    gemm_wmma<NQ, 32, true, false, false><<<dim3(32 / 32, NQ / 128, B_), blk, 0, stream>>>(
        x, (long)C_ * NQ, Wa, ba, nullptr, 0, attn, (long)NQ * 32, NQ);
    // 4) deformable sampling + attention combine
    const long tot = (long)B_ * NQ * C_;
    deform_sample<<<dim3((unsigned)(tot / 256)), blk, 0, stream>>>(v, offb, attn, outp);
    // 5) output projection + bias + residual, transposed store to (B,C,H,W)
    gemm_wmma<C_, C_, false, true, true><<<dim3(C_ / 32, NQ / 128, B_), blk, 0, stream>>>(
        outp, (long)NQ * C_, Wout, bout, x, (long)C_ * NQ, out, (long)C_ * NQ, NQ);
}